// GCNEncoder_64287070486608
// MI455X (gfx1250) — compile-verified
//
#include <hip/hip_runtime.h>
#include <hip/hip_bf16.h>

// GCN 2-layer forward for MI455X (gfx1250, wave32).
//  - GEMMs use V_WMMA_F32_16X16X4_F32 (exact fp32, matches reference math).
//  - Edge aggregation uses global f32 atomics (L2-resident: 5.12MB features << 192MB L2).
//  - Bias folded into scatter-destination initialization.
//
// Workspace layout (d_ws):
//   [0, N)                  float  dinv  (also used as deg accumulator)
//   [N, N + N*128)          float  bufH  (GEMM output / messages)
//   [N + N*128, N + 2*N*128) float bufA  (aggregation target layer 1)
// Requires ws_size >= (N + 2*N*128)*4 bytes  (~10.3 MB for N=10000).

typedef float v2f __attribute__((ext_vector_type(2)));
typedef float v8f __attribute__((ext_vector_type(8)));

#define DFEAT 128

// ---------------------------------------------------------------- utilities
__global__ __launch_bounds__(256) void k_zero_f32(float* __restrict__ p, int n) {
    int i = blockIdx.x * 256 + threadIdx.x;
    if (i < n) p[i] = 0.0f;
}

// deg[dst] += 1 for every real edge (self loops handled analytically later)
__global__ __launch_bounds__(256) void k_count_deg(const int* __restrict__ dst,
                                                   float* __restrict__ deg, int E) {
    int i = blockIdx.x * 256 + threadIdx.x;
    if (i < E) atomicAdd(&deg[dst[i]], 1.0f);
}

// dinv[i] = rsqrt(count + 1)   (+1 == self loop; always > 0)
__global__ __launch_bounds__(256) void k_finalize_dinv(float* __restrict__ deg, int N) {
    int i = blockIdx.x * 256 + threadIdx.x;
    if (i < N) deg[i] = rsqrtf(deg[i] + 1.0f);
}

// out[n, f] = bias[f]  (bias folded into the scatter-add destination)
__global__ __launch_bounds__(256) void k_init_bias(float* __restrict__ out,
                                                   const float* __restrict__ b, int total) {
    int i = blockIdx.x * 256 + threadIdx.x;
    if (i < total) out[i] = b[i & (DFEAT - 1)];
}

// ---------------------------------------------------------------- fp32 WMMA GEMM
// H[N,128] = X[N,128] @ W[128,128]
// Block = 256 threads = 8 waves; block b handles rows [16b,16b+16); wave w handles
// output columns [16w, 16w+16). Each wave: 32 chained V_WMMA_F32_16X16X4_F32.
//
// A fragment (16x4 fp32, 2 VGPRs/lane):
//   lane = 16*half + idx  ->  holds A[idx][k0+2*half] in a.x, A[idx][k0+2*half+1] in a.y
// B fragment (4x16 fp32, 2 VGPRs/lane):
//   holds B[k0+2*half][col0+idx] in b.x, B[k0+2*half+1][col0+idx] in b.y
// C/D (16x16 fp32, 8 VGPRs/lane): acc[r] = D[r + 8*half][col0+idx]
__global__ __launch_bounds__(256) void k_gemm_wmma_f32(const float* __restrict__ X,
                                                       const float* __restrict__ W,
                                                       float* __restrict__ H, int N) {
    const int wave = threadIdx.x >> 5;   // 0..7 -> column tile
    const int lane = threadIdx.x & 31;
    const int half = lane >> 4;          // 0 or 1
    const int idx  = lane & 15;
    const int row0 = blockIdx.x << 4;
    const int col0 = wave << 4;

    // clamp load row (no divergence: EXEC must stay all-ones across the WMMAs)
    int lrow = row0 + idx;
    if (lrow > N - 1) lrow = N - 1;
    const float*  xrow = X + (size_t)lrow * DFEAT + 2 * half;
    const float*  wcol = W + (size_t)(2 * half) * DFEAT + col0 + idx;

    v8f acc = {};
#pragma unroll
    for (int k = 0; k < DFEAT; k += 4) {
        const float2 av = *(const float2*)(xrow + k);   // global_load_b64
        v2f a; a.x = av.x; a.y = av.y;
        v2f b;
        b.x = wcol[(size_t)k * DFEAT];
        b.y = wcol[(size_t)k * DFEAT + DFEAT];
        acc = __builtin_amdgcn_wmma_f32_16x16x4_f32(
            /*neg_a=*/false, a, /*neg_b=*/false, b,
            /*c_mod=*/(short)0, acc, /*reuse_a=*/false, /*reuse_b=*/false);
    }

    float* orow = H + (size_t)(row0 + half * 8) * DFEAT + col0 + idx;
#pragma unroll
    for (int r = 0; r < 8; ++r) {
        if (row0 + half * 8 + r < N) orow[(size_t)r * DFEAT] = acc[r];
    }
}

// ---------------------------------------------------------------- edge scatter
// One wave per (edge or self-loop); lane handles 4 consecutive features.
// Out[dst] += H[src] * dinv[src]*dinv[dst]   via non-returning f32 atomics.
__global__ __launch_bounds__(256) void k_aggregate(const float* __restrict__ H,
                                                   const int* __restrict__ src,
                                                   const int* __restrict__ dst,
                                                   const float* __restrict__ dinv,
                                                   float* __restrict__ Out,
                                                   int E, int N) {
    const int e    = blockIdx.x * 8 + (threadIdx.x >> 5);
    const int lane = threadIdx.x & 31;
    if (e >= E + N) return;

    int s, d;
    if (e < E) { s = src[e]; d = dst[e]; }
    else       { s = e - E; d = s; }                 // self loop
    const float nrm = dinv[s] * dinv[d];

    const float4 v = *(const float4*)(H + (size_t)s * DFEAT + lane * 4);  // b128 gather
    float* o = Out + (size_t)d * DFEAT + lane * 4;
    atomicAdd(o + 0, v.x * nrm);   // global_atomic_add_f32, no return
    atomicAdd(o + 1, v.y * nrm);
    atomicAdd(o + 2, v.z * nrm);
    atomicAdd(o + 3, v.w * nrm);
}

// ---------------------------------------------------------------- driver
extern "C" void kernel_launch(void* const* d_in, const int* in_sizes, int n_in,
                              void* d_out, int out_size, void* d_ws, size_t ws_size,
                              hipStream_t stream) {
    const float* x     = (const float*)d_in[0];   // [N,128]
    const int*   edges = (const int*)  d_in[1];   // [2,E] row-major
    const float* W1    = (const float*)d_in[2];
    const float* b1    = (const float*)d_in[3];
    const float* W2    = (const float*)d_in[4];
    const float* b2    = (const float*)d_in[5];
    float*       out   = (float*)d_out;

    const int N = in_sizes[0] / DFEAT;
    const int E = in_sizes[1] / 2;
    const int* srcI = edges;
    const int* dstI = edges + E;

    float* dinv = (float*)d_ws;
    float* bufH = dinv + N;                     // N*128
    float* bufA = bufH + (size_t)N * DFEAT;     // N*128

    const int NF       = N * DFEAT;
    const int gN       = (N + 255) / 256;
    const int gE       = (E + 255) / 256;
    const int gNF      = (NF + 255) / 256;
    const int gTiles   = (N + 15) / 16;
    const int gEdges   = (E + N + 7) / 8;       // 8 edges per 256-thread block

    // ---- normalization
    k_zero_f32     <<<gN,     256, 0, stream>>>(dinv, N);
    k_count_deg    <<<gE,     256, 0, stream>>>(dstI, dinv, E);
    k_finalize_dinv<<<gN,     256, 0, stream>>>(dinv, N);

    // ---- layer 1: H = X @ W1 ; A = b1 + scatter(H)
    k_gemm_wmma_f32<<<gTiles, 256, 0, stream>>>(x, W1, bufH, N);
    k_init_bias    <<<gNF,    256, 0, stream>>>(bufA, b1, NF);
    k_aggregate    <<<gEdges, 256, 0, stream>>>(bufH, srcI, dstI, dinv, bufA, E, N);

    // ---- layer 2: H = A @ W2 ; out = b2 + scatter(H)
    k_gemm_wmma_f32<<<gTiles, 256, 0, stream>>>(bufA, W2, bufH, N);
    k_init_bias    <<<gNF,    256, 0, stream>>>(out, b2, NF);
    k_aggregate    <<<gEdges, 256, 0, stream>>>(bufH, srcI, dstI, dinv, out, E, N);
}